// DeepHitLoss_25125558682256
// MI455X (gfx1250) — compile-verified
//
#include <hip/hip_runtime.h>
#include <math.h>

typedef float v2f __attribute__((ext_vector_type(2)));
typedef float v8f __attribute__((ext_vector_type(8)));

#define ALPHA  0.5f
#define EPS    1e-8f
#define MARGIN 0.1f
#define JTILE  1024

// ---------------------------------------------------------------------------
// Kernel 0: zero the accumulators in workspace.
// acc[0] = log-likelihood sum (float)
// acc[1] = ranking pair-loss sum (float)
// acc[2] = pair count (unsigned int bits)
// ---------------------------------------------------------------------------
__global__ void dh_init_acc(float* acc) {
    acc[0] = 0.0f;
    acc[1] = 0.0f;
    reinterpret_cast<unsigned int*>(acc)[2] = 0u;
}

// ---------------------------------------------------------------------------
// Kernel 1: risk[b] = sum_t h[b,t] via V_WMMA_F32_16X16X4_F32 with B = ones.
// One wave32 handles 16 rows; 64 chained WMMAs accumulate K=256 into f32 C.
//
// A-layout (16x4 f32): lanes 0-15 -> M=lane, v0=K0, v1=K1;
//                      lanes 16-31 -> M=lane-16, v0=K2, v1=K3.
// Each lane loads a contiguous float2 at column (kk + 2*(lane>=16)).
// D-layout (16x16 f32): lane 0 holds rows 0..7 (col 0) in c[0..7],
//                       lane 16 holds rows 8..15 (col 0) in c[0..7].
// ---------------------------------------------------------------------------
__global__ __launch_bounds__(256) void dh_risk_wmma(const float* __restrict__ h,
                                                    float* __restrict__ risk,
                                                    int T) {
    const int wave    = threadIdx.x >> 5;
    const int lane    = threadIdx.x & 31;
    const int rowBase = (blockIdx.x * 8 + wave) * 16;
    const int r       = lane & 15;
    const int koff    = (lane >> 4) << 1;   // 0 for lanes 0-15, 2 for lanes 16-31
    const float* rowPtr = h + (size_t)(rowBase + r) * (size_t)T;

    v8f c = {};                                   // f32 accumulator, 8 VGPRs
    v2f bones; bones[0] = 1.0f; bones[1] = 1.0f;  // B = ones (layout-agnostic)

#pragma unroll 8
    for (int kk = 0; kk < T; kk += 4) {
        v2f a = *(const v2f*)(rowPtr + kk + koff);  // global_load_b64
        c = __builtin_amdgcn_wmma_f32_16x16x4_f32(false, a, false, bones,
                                                  (short)0, c, false, false);
    }

    if (lane == 0) {
#pragma unroll
        for (int i = 0; i < 8; ++i) risk[rowBase + i] = c[i];
    } else if (lane == 16) {
#pragma unroll
        for (int i = 0; i < 8; ++i) risk[rowBase + 8 + i] = c[i];
    }
}

// ---------------------------------------------------------------------------
// Kernel 2: per-row log-likelihood.  blockDim.x == T (256 = 8 waves).
// P = prod_{i<t}(1-h_i); prob = event ? P*h_t : P*(1-h_t).
// ---------------------------------------------------------------------------
__global__ __launch_bounds__(256) void dh_loglik(const float* __restrict__ h,
                                                 const int* __restrict__ times,
                                                 const int* __restrict__ events,
                                                 float* __restrict__ acc,
                                                 int T) {
    const int b   = blockIdx.x;
    const int tid = threadIdx.x;

    int t = times[b];
    t = t < 0 ? 0 : (t > T - 1 ? T - 1 : t);

    const float hv = h[(size_t)b * (size_t)T + tid];
    float contrib  = (tid < t) ? (1.0f - hv) : 1.0f;

    __shared__ float sprod[32];
    __shared__ float sht;
    if (tid == t) sht = hv;     // exactly one writer (t in [0, T-1])

    // wave32 product reduction
    for (int off = 16; off > 0; off >>= 1)
        contrib *= __shfl_xor(contrib, off, 32);
    if ((tid & 31) == 0) sprod[tid >> 5] = contrib;
    __syncthreads();

    if (tid == 0) {
        const int nWaves = blockDim.x >> 5;
        float P = 1.0f;
        for (int i = 0; i < nWaves; ++i) P *= sprod[i];
        const float ht   = sht;
        const float prob = (events[b] == 1) ? (P * ht) : (P * (1.0f - ht));
        atomicAdd(&acc[0], -logf(prob + EPS));
    }
}

// ---------------------------------------------------------------------------
// Kernel 3: ranking pair loss, j-tiled through LDS.
// Grid: (B/256 i-tiles, ceil(B/JTILE) j-chunks).  Each block stages a
// JTILE-wide slice of (times, risk) in LDS; each thread owns one i and scans
// the LDS slice for j > i.
// mask(i,j) = events[i]==1 && times[i] < times[j]
// loss(i,j) = relu(risk[j] - risk[i] + MARGIN)
// ---------------------------------------------------------------------------
__global__ __launch_bounds__(256) void dh_ranking(const float* __restrict__ risk,
                                                  const int* __restrict__ times,
                                                  const int* __restrict__ events,
                                                  float* __restrict__ acc,
                                                  int B) {
    __shared__ float s_risk[JTILE];
    __shared__ int   s_time[JTILE];

    const int i     = blockIdx.x * blockDim.x + threadIdx.x;
    const int jBase = blockIdx.y * JTILE;

    // cooperative stage of the j-chunk into LDS
    for (int k = threadIdx.x; k < JTILE; k += blockDim.x) {
        const int j = jBase + k;
        if (j < B) {
            s_risk[k] = risk[j];
            s_time[k] = times[j];
        }
    }
    __syncthreads();

    float        s   = 0.0f;
    unsigned int cnt = 0u;

    const bool active = (i < B) && (events[i] == 1);
    if (active) {
        const float ri = risk[i];
        const int   ti = times[i];
        int kStart = i + 1 - jBase; if (kStart < 0) kStart = 0;
        int kEnd   = B - jBase;     if (kEnd > JTILE) kEnd = JTILE;
        for (int k = kStart; k < kEnd; ++k) {
            if (ti < s_time[k]) {
                float pl = s_risk[k] - ri + MARGIN;
                s   += (pl > 0.0f) ? pl : 0.0f;
                cnt += 1u;
            }
        }
    }

    // wave32 reduction, then one atomic per wave leader
    for (int off = 16; off > 0; off >>= 1) {
        s   += __shfl_xor(s, off, 32);
        cnt += (unsigned int)__shfl_xor((int)cnt, off, 32);
    }
    if ((threadIdx.x & 31) == 0) {
        atomicAdd(&acc[1], s);
        atomicAdd(reinterpret_cast<unsigned int*>(acc) + 2, cnt);
    }
}

// ---------------------------------------------------------------------------
// Kernel 4: combine: out = ll_sum/B + ALPHA * (n>0 ? pair_sum/n : 0)
// ---------------------------------------------------------------------------
__global__ void dh_finalize(const float* __restrict__ acc,
                            float* __restrict__ out, int B) {
    const float        ll = acc[0] / (float)B;
    const unsigned int n  = reinterpret_cast<const unsigned int*>(acc)[2];
    const float        rk = (n > 0u) ? (acc[1] / (float)n) : 0.0f;
    out[0] = ll + ALPHA * rk;
}

// ---------------------------------------------------------------------------
extern "C" void kernel_launch(void* const* d_in, const int* in_sizes, int n_in,
                              void* d_out, int out_size, void* d_ws, size_t ws_size,
                              hipStream_t stream) {
    const float* h      = (const float*)d_in[0];   // [B, T] f32
    const int*   times  = (const int*)d_in[1];     // [B] int
    const int*   events = (const int*)d_in[2];     // [B] int

    const int B = in_sizes[1];
    const int T = in_sizes[0] / B;

    float* ws   = (float*)d_ws;
    float* risk = ws;          // B floats
    float* acc  = ws + B;      // [ll_sum, pair_sum, (uint)pair_count]
    float* out  = (float*)d_out;

    dh_init_acc<<<1, 1, 0, stream>>>(acc);
    dh_risk_wmma<<<B / 128, 256, 0, stream>>>(h, risk, T);   // 8 waves x 16 rows
    dh_loglik<<<B, T, 0, stream>>>(h, times, events, acc, T);

    dim3 rgrid((B + 255) / 256, (B + JTILE - 1) / JTILE);
    dh_ranking<<<rgrid, 256, 0, stream>>>(risk, times, events, acc, B);

    dh_finalize<<<1, 1, 0, stream>>>(acc, out, B);
}